// CompactHash_46952582479956
// MI455X (gfx1250) — compile-verified
//
#include <hip/hip_runtime.h>
#include <cstdint>

#define BLOCK 256

// One level of the multiresolution compact-hash encoding for one sample.
// USE_LDS=true  -> embedding gathers come from the LDS-staged level-0 table
//                  (padded: 16-row block t at dword 36*t) via ds_load_b128.
// USE_LDS=false -> embedding gathers are global_load_b128 from the L2-resident
//                  table slice starting at `off`.
template <bool USE_LDS>
__device__ __forceinline__ void level_encode(
    const int L, const unsigned pmask, const unsigned off,
    const float px, const float py, const float pz,
    const float* __restrict__ embeddings,
    const float* __restrict__ codebook,
    const float* s_emb0,
    float* __restrict__ orow)
{
    const float res = (float)(16 << L);

    const float x = px * res, y = py * res, z = pz * res;
    const float fx = floorf(x), fy = floorf(y), fz = floorf(z);
    const float tx = x - fx, ty = y - fy, tz = z - fz;
    const float mx = 1.f - tx, my = 1.f - ty, mz = 1.f - tz;
    const unsigned ux = (unsigned)(int)fx;
    const unsigned uy = (unsigned)(int)fy;
    const unsigned uz = (unsigned)(int)fz;

    // hash components for both corner choices per dim (PRIMES1 / PRIMES2)
    const unsigned ax0 = ux,               ax1 = ux + 1u;
    const unsigned ay0 = uy * 2654435761u, ay1 = (uy + 1u) * 2654435761u;
    const unsigned az0 = uz * 805459861u,  az1 = (uz + 1u) * 805459861u;
    const unsigned by0 = uy * 2654435767u, by1 = (uy + 1u) * 2654435767u;
    const unsigned bz0 = uz * 805459871u,  bz1 = (uz + 1u) * 805459871u;

    float o0 = 0.f, o1 = 0.f;
#pragma unroll 1
    for (int c = 0; c < 8; ++c) {
        const bool b0 = (c & 1) != 0, b1 = (c & 2) != 0, b2 = (c & 4) != 0;
        const unsigned h1 =
            ((b0 ? ax1 : ax0) ^ (b1 ? ay1 : ay0) ^ (b2 ? az1 : az0)) & pmask;
        const unsigned h2 =
            ((b0 ? ax1 : ax0) ^ (b1 ? by1 : by0) ^ (b2 ? bz1 : bz0)) & 16383u;

        // codebook row: 16 f32 = 4x global_load_b128, 64B aligned
        const float4* cb =
            (const float4*)(codebook + (size_t)((((unsigned)L << 14) + h2) << 4));
        const float4 c0 = cb[0], c1 = cb[1], c2 = cb[2], c3 = cb[3];

        const float e0  = __expf(c0.x), e1  = __expf(c0.y);
        const float e2  = __expf(c0.z), e3  = __expf(c0.w);
        const float e4  = __expf(c1.x), e5  = __expf(c1.y);
        const float e6  = __expf(c1.z), e7  = __expf(c1.w);
        const float e8  = __expf(c2.x), e9  = __expf(c2.y);
        const float e10 = __expf(c2.z), e11 = __expf(c2.w);
        const float e12 = __expf(c3.x), e13 = __expf(c3.y);
        const float e14 = __expf(c3.z), e15 = __expf(c3.w);

        const float ssum =
            (((e0 + e1) + (e2 + e3)) + ((e4 + e5) + (e6 + e7))) +
            (((e8 + e9) + (e10 + e11)) + ((e12 + e13) + (e14 + e15)));

        // 16 consecutive embedding rows (window never wraps): 128 B contiguous
        float4 v0, v1, v2, v3, v4, v5, v6, v7;
        if constexpr (USE_LDS) {
            const float4* eb = (const float4*)&s_emb0[36u * (h1 & 255u)];
            v0 = eb[0]; v1 = eb[1]; v2 = eb[2]; v3 = eb[3];
            v4 = eb[4]; v5 = eb[5]; v6 = eb[6]; v7 = eb[7];
        } else {
            const unsigned base = (h1 * 16u) & pmask;
            const float4* eb =
                (const float4*)(embeddings + (size_t)(off + base) * 2u);
            v0 = eb[0]; v1 = eb[1]; v2 = eb[2]; v3 = eb[3];
            v4 = eb[4]; v5 = eb[5]; v6 = eb[6]; v7 = eb[7];
        }

        const float f0 =
            e0 * v0.x + e1 * v0.z + e2  * v1.x + e3  * v1.z +
            e4 * v2.x + e5 * v2.z + e6  * v3.x + e7  * v3.z +
            e8 * v4.x + e9 * v4.z + e10 * v5.x + e11 * v5.z +
            e12 * v6.x + e13 * v6.z + e14 * v7.x + e15 * v7.z;
        const float f1 =
            e0 * v0.y + e1 * v0.w + e2  * v1.y + e3  * v1.w +
            e4 * v2.y + e5 * v2.w + e6  * v3.y + e7  * v3.w +
            e8 * v4.y + e9 * v4.w + e10 * v5.y + e11 * v5.w +
            e12 * v6.y + e13 * v6.w + e14 * v7.y + e15 * v7.w;

        const float wc = (b0 ? tx : mx) * (b1 ? ty : my) * (b2 ? tz : mz);
        const float sc = wc * __builtin_amdgcn_rcpf(ssum);
        o0 += f0 * sc;
        o1 += f1 * sc;
    }
    orow[2 * L + 0] = o0;
    orow[2 * L + 1] = o1;
}

__global__ __launch_bounds__(BLOCK) void hashenc_kernel(
    const float* __restrict__ inputs,
    const float* __restrict__ embeddings,
    const float* __restrict__ codebook,
    float* __restrict__ out,
    int batch)
{
    // padded layout: block t (16 rows = 32 floats) lives at dword 36*t
    // -> 16B aligned, start-bank stride 36 (mod 64) breaks conflicts.
    __shared__ float s_emb0[256 * 36];  // 36 KB

    // ---- async stage level-0 embeddings (4096 rows x 2 f32 = 32 KB) ----
    {
        const unsigned t = threadIdx.x;
        const unsigned long long g0 =
            (unsigned long long)(uintptr_t)(embeddings + 32u * t);
        const unsigned l0 = (unsigned)(uintptr_t)(&s_emb0[36u * t]);
#pragma unroll
        for (int k = 0; k < 8; ++k) {
            unsigned long long g = g0 + 16ull * (unsigned)k;
            unsigned l = l0 + 16u * (unsigned)k;
            asm volatile("global_load_async_to_lds_b128 %0, %1, off"
                         :: "v"(l), "v"(g) : "memory");
        }
    }
    asm volatile("s_wait_asynccnt 0" ::: "memory");
    __syncthreads();

    const int s = blockIdx.x * BLOCK + (int)threadIdx.x;
    if (s >= batch) return;

    const float px = inputs[3 * s + 0];
    const float py = inputs[3 * s + 1];
    const float pz = inputs[3 * s + 2];

    float* orow = out + (size_t)s * 16;

    // level 0: params = 4096, table staged in LDS
    level_encode<true>(0, 4095u, 0u, px, py, pz,
                       embeddings, codebook, s_emb0, orow);

    // levels 1..7: params = min(2^19, 2^(12+3L)); global (L2-resident) gathers
    unsigned off = 4096u;
#pragma unroll 1
    for (int L = 1; L < 8; ++L) {
        const unsigned sh = (L < 3) ? (12u + 3u * (unsigned)L) : 19u;
        const unsigned pmask = (1u << sh) - 1u;
        level_encode<false>(L, pmask, off, px, py, pz,
                            embeddings, codebook, s_emb0, orow);
        off += pmask + 1u;
    }
}

extern "C" void kernel_launch(void* const* d_in, const int* in_sizes, int n_in,
                              void* d_out, int out_size, void* d_ws, size_t ws_size,
                              hipStream_t stream) {
    const float* inputs     = (const float*)d_in[0];
    const float* embeddings = (const float*)d_in[1];
    const float* codebook   = (const float*)d_in[2];
    float* out = (float*)d_out;

    const int batch = in_sizes[0] / 3;
    const int grid = (batch + BLOCK - 1) / BLOCK;
    hipLaunchKernelGGL(hashenc_kernel, dim3(grid), dim3(BLOCK), 0, stream,
                       inputs, embeddings, codebook, out, batch);
}